// PPM_45062796870410
// MI455X (gfx1250) — compile-verified
//
#include <hip/hip_runtime.h>
#include <cstdint>

// Problem dimensions (fixed by the reference)
#define E_  5
#define MB_ 2048
#define N_  1024
#define L_  10
#define PROBE_CTS_ 1.0e5f
#define VOXEL_CM_  (0.01f / 1024.0f)

typedef __attribute__((ext_vector_type(2))) float v2f;
typedef __attribute__((ext_vector_type(8))) float v8f;
typedef int v4i_ __attribute__((vector_size(16)));   // matches async builtin param

#define AS1 __attribute__((address_space(1)))
#define AS3 __attribute__((address_space(3)))

// One block (256 threads = 8 wave32) per image row m. Each thread owns 4
// consecutive voxels n = 4*tid .. 4*tid+3 so all VMEM traffic is fully
// coalesced b128 streams.
__global__ __launch_bounds__(256)
void PPM_45062796870410_kernel(const float* __restrict__ xp,      // (E,MB,N)
                               const float* __restrict__ attCS,   // (E,)
                               const float* __restrict__ flu,     // (L,)
                               const float* __restrict__ SA,      // (L,V)
                               const int*   __restrict__ lte,     // (L,)
                               float*       __restrict__ out)     // L*MB fl + MB trans
{
    __shared__ float xp_lds[E_][N_];   // 20 KB: xp row staged once, read twice
    __shared__ float wsum[8];          // per-wave scan totals
    __shared__ float P[16][8];         // (quantity x wave) partials for WMMA reduce

    const int tid  = threadIdx.x;
    const int lane = tid & 31;
    const int wid  = tid >> 5;
    const int m    = blockIdx.x;
    const int t4   = tid * 4;

    // zero the WMMA partial matrix early (covered by the first barrier)
    if (tid < 128) ((float*)P)[tid] = 0.0f;

    // ---- Phase 1: stage xp[:, m, :] into LDS with async b128 copies --------
    #pragma unroll
    for (int e = 0; e < E_; ++e) {
        const float* g = xp + ((size_t)e * MB_ + m) * N_ + t4;
#if __has_builtin(__builtin_amdgcn_global_load_async_to_lds_b128)
        __builtin_amdgcn_global_load_async_to_lds_b128(
            (AS1 v4i_*)g, (AS3 v4i_*)&xp_lds[e][t4], 0, 0);
#else
        *(float4*)&xp_lds[e][t4] = *(const float4*)g;
#endif
    }

    // Prefetch the 10 SA_theta streams for phase 3 while the scan runs.
    #pragma unroll
    for (int l = 0; l < L_; ++l)
        __builtin_prefetch(SA + (size_t)l * ((size_t)MB_ * N_) + (size_t)m * N_ + t4, 0, 0);

#if __has_builtin(__builtin_amdgcn_s_wait_asynccnt)
    __builtin_amdgcn_s_wait_asynccnt(0);
#else
    asm volatile("s_wait_asynccnt 0x0" ::: "memory");
#endif
    __syncthreads();

    // ---- Phase 2: lac values + block-wide exclusive prefix scan ------------
    const float a0 = attCS[0], a1 = attCS[1], a2 = attCS[2], a3 = attCS[3], a4 = attCS[4];
    float v0 = a0*xp_lds[0][t4+0] + a1*xp_lds[1][t4+0] + a2*xp_lds[2][t4+0] + a3*xp_lds[3][t4+0] + a4*xp_lds[4][t4+0];
    float v1 = a0*xp_lds[0][t4+1] + a1*xp_lds[1][t4+1] + a2*xp_lds[2][t4+1] + a3*xp_lds[3][t4+1] + a4*xp_lds[4][t4+1];
    float v2 = a0*xp_lds[0][t4+2] + a1*xp_lds[1][t4+2] + a2*xp_lds[2][t4+2] + a3*xp_lds[3][t4+2] + a4*xp_lds[4][t4+2];
    float v3 = a0*xp_lds[0][t4+3] + a1*xp_lds[1][t4+3] + a2*xp_lds[2][t4+3] + a3*xp_lds[3][t4+3] + a4*xp_lds[4][t4+3];
    const float s01 = v0 + v1, s012 = s01 + v2, s = s012 + v3;

    // intra-wave inclusive scan of per-thread sums (wave32)
    float incl = s;
    #pragma unroll
    for (int d = 1; d < 32; d <<= 1) {
        float tt = __shfl_up(incl, d, 32);
        if (lane >= d) incl += tt;
    }
    if (lane == 31) wsum[wid] = incl;
    __syncthreads();

    float wbase = 0.0f, total = 0.0f;
    #pragma unroll
    for (int w = 0; w < 8; ++w) {
        float ww = wsum[w];
        total += ww;
        if (w < wid) wbase += ww;
    }
    const float ex = wbase + (incl - s);   // exclusive prefix at voxel t4

    // attenuation entering each of this thread's 4 voxels (shifted prefix)
    const float pa0 = PROBE_CTS_ * __expf(-VOXEL_CM_ * ex);
    const float pa1 = PROBE_CTS_ * __expf(-VOXEL_CM_ * (ex + v0));
    const float pa2 = PROBE_CTS_ * __expf(-VOXEL_CM_ * (ex + s01));
    const float pa3 = PROBE_CTS_ * __expf(-VOXEL_CM_ * (ex + s012));

    if (tid == 0)
        out[L_ * MB_ + m] = PROBE_CTS_ * __expf(-VOXEL_CM_ * total);  // transmission

    // ---- Phase 3: fluorescence accumulation (streams SA_theta once) --------
    float acc[L_];
    #pragma unroll
    for (int l = 0; l < L_; ++l) {
        const int e = lte[l];
        const float4 sa = *(const float4*)(SA + (size_t)l * ((size_t)MB_ * N_) + (size_t)m * N_ + t4);
        const float* xr = &xp_lds[e][t4];
        acc[l] = (pa0 * xr[0] * sa.x + pa1 * xr[1] * sa.y +
                  pa2 * xr[2] * sa.z + pa3 * xr[3] * sa.w) * flu[l];
    }

    // intra-wave butterfly reduction; wave leaders deposit partials in P
    #pragma unroll
    for (int l = 0; l < L_; ++l) {
        float x = acc[l];
        #pragma unroll
        for (int d = 16; d > 0; d >>= 1) x += __shfl_xor(x, d, 32);
        if (lane == 0) P[l][wid] = x;
    }
    __syncthreads();

    // ---- Final cross-wave reduction on the matrix pipe ---------------------
    // D(16x16) = A(16x4) * ones(4x16) + C  collapses 4 wave-columns per op;
    // two chained v_wmma_f32_16x16x4_f32 sum all 8 waves per quantity.
#if __has_builtin(__builtin_amdgcn_wmma_f32_16x16x4_f32)
    {
        const int half = lane >> 4;      // lanes 16-31 supply K=2,3 per A layout
        const int r    = lane & 15;      // matrix row M
        v2f A0, A1, B;
        A0[0] = P[r][half * 2 + 0];  A0[1] = P[r][half * 2 + 1];
        A1[0] = P[r][half * 2 + 4];  A1[1] = P[r][half * 2 + 5];
        B[0] = 1.0f;                 B[1] = 1.0f;
        v8f c = {};
        c = __builtin_amdgcn_wmma_f32_16x16x4_f32(false, A0, false, B, (short)0, c, false, false);
        c = __builtin_amdgcn_wmma_f32_16x16x4_f32(false, A1, false, B, (short)0, c, false, false);
        // D layout: VGPR q, lanes 0-15 -> row q ; lanes 16-31 -> row q+8
        if (wid == 0) {
            if (lane == 0) {
                #pragma unroll
                for (int q = 0; q < 8; ++q) out[q * MB_ + m] = c[q];
            } else if (lane == 16) {
                out[8 * MB_ + m] = c[0];
                out[9 * MB_ + m] = c[1];
            }
        }
    }
#else
    if (wid == 0 && lane < L_) {
        float sum = 0.0f;
        #pragma unroll
        for (int w = 0; w < 8; ++w) sum += P[lane][w];
        out[lane * MB_ + m] = sum;
    }
#endif
}

extern "C" void kernel_launch(void* const* d_in, const int* in_sizes, int n_in,
                              void* d_out, int out_size, void* d_ws, size_t ws_size,
                              hipStream_t stream) {
    const float* xp    = (const float*)d_in[0];
    const float* attCS = (const float*)d_in[1];
    const float* flu   = (const float*)d_in[2];
    const float* SA    = (const float*)d_in[3];
    const int*   lte   = (const int*)d_in[4];
    float*       out   = (float*)d_out;

    PPM_45062796870410_kernel<<<MB_, 256, 0, stream>>>(xp, attCS, flu, SA, lte, out);

    (void)in_sizes; (void)n_in; (void)out_size; (void)d_ws; (void)ws_size;
}